// MMDLoss_64098091925653
// MI455X (gfx1250) — compile-verified
//
#include <hip/hip_runtime.h>

typedef __attribute__((ext_vector_type(2))) float v2f;
typedef __attribute__((ext_vector_type(4))) float v4f;
typedef __attribute__((ext_vector_type(8))) float v8f;

#define N_HALF 4096
#define NTOT   8192
#define DDIM   512
#define BM     128     // C tile is BM x BM
#define KC     32      // K chunk staged through LDS
#define LDST   36      // padded LDS row stride in floats -> conflict-free frag reads
#define NCHUNK (DDIM / KC)
#define BUFELTS (BM * LDST)

// LDS byte offset of a __shared__ object (for async-to-LDS addressing)
__device__ __forceinline__ unsigned as_lds_off(const void* p) {
  return (unsigned)(unsigned long long)(__attribute__((address_space(3))) const void*)p;
}

// ---------------- pass 1a: per-row squared norms (one wave per row) ----------
__global__ __launch_bounds__(256)
void mmd_rowsq(const float* __restrict__ src, const float* __restrict__ tgt,
               float* __restrict__ sq) {
  const int wave = threadIdx.x >> 5, lane = threadIdx.x & 31;
  const int row = blockIdx.x * 8 + wave;            // 1024 blocks x 8 waves = 8192 rows
  const float* base = (row < N_HALF) ? src + (size_t)row * DDIM
                                     : tgt + (size_t)(row - N_HALF) * DDIM;
  float s = 0.f;
  for (int c = lane; c < DDIM; c += 32) { float v = base[c]; s += v * v; }
  #pragma unroll
  for (int off = 16; off; off >>= 1) s += __shfl_xor(s, off, 32);
  if (lane == 0) sq[row] = s;
}

// ---------------- pass 1b: column sums (coalesced, deterministic) ------------
__global__ __launch_bounds__(256)
void mmd_colsum(const float* __restrict__ src, const float* __restrict__ tgt,
                float* __restrict__ colsum) {
  const int c = blockIdx.x * 256 + threadIdx.x;     // 2 blocks -> 512 columns
  float s = 0.f;
  for (int r = 0; r < N_HALF; ++r) s += src[(size_t)r * DDIM + c];
  for (int r = 0; r < N_HALF; ++r) s += tgt[(size_t)r * DDIM + c];
  colsum[c] = s;
}

// ---------------- pass 2: bandwidth -> gmin = 1/(16*bw) ----------------------
__global__ __launch_bounds__(256)
void mmd_bandwidth(const float* __restrict__ sq, const float* __restrict__ colsum,
                   float* __restrict__ gmin_out) {
  __shared__ float red[256];
  const int t = threadIdx.x;
  float a = 0.f;
  for (int i = t; i < NTOT; i += 256) a += sq[i];
  float b = 0.f;
  for (int i = t; i < DDIM; i += 256) { float v = colsum[i]; b += v * v; }
  red[t] = a; __syncthreads();
  for (int off = 128; off; off >>= 1) { if (t < off) red[t] += red[t + off]; __syncthreads(); }
  float sumsq = red[0]; __syncthreads();
  red[t] = b; __syncthreads();
  for (int off = 128; off; off >>= 1) { if (t < off) red[t] += red[t + off]; __syncthreads(); }
  if (t == 0) {
    float s2 = red[0];
    const float n = (float)NTOT;
    float sumL2 = 2.f * n * sumsq - 2.f * s2;            // closed-form sum(L2)
    float bw = sumL2 / (n * n - n);
    bw = bw * 0.25f;                                     // / KERNEL_MUL^(KERNEL_NUM//2) = /4
    gmin_out[0] = 1.f / (bw * 16.f);                     // smallest gamma; others via squaring
  }
}

// ---------------- pass 3: fused Gram GEMM + 5-kernel RBF + tile reduction ----
// Symmetric: only by <= bx tiles do work (transpose tile has identical sum+sign).
// Double-buffered LDS, filled via CDNA5 async load-to-LDS (ASYNCcnt-tracked).
__global__ __launch_bounds__(256)
void mmd_gram(const float* __restrict__ src, const float* __restrict__ tgt,
              const float* __restrict__ sq, const float* __restrict__ gptr,
              float* __restrict__ partial) {
  __shared__ float lA[2][BUFELTS];
  __shared__ float lB[2][BUFELTS];
  __shared__ float sqI[BM];
  __shared__ float sqJ[BM];
  __shared__ float redw[8];

  const int bx = blockIdx.x & 63;               // j tile
  const int by = blockIdx.x >> 6;               // i tile
  if (by > bx) return;                          // symmetry: lower triangle skipped
  const int i0 = by * BM, j0 = bx * BM;
  const float* abase = (i0 < N_HALF) ? src + (size_t)i0 * DDIM
                                     : tgt + (size_t)(i0 - N_HALF) * DDIM;
  const float* bbase = (j0 < N_HALF) ? src + (size_t)j0 * DDIM
                                     : tgt + (size_t)(j0 - N_HALF) * DDIM;

  const int t    = threadIdx.x;
  const int lane = t & 31, wave = t >> 5;
  const int m0   = (wave & 3) * 32;             // wave's 32-row patch
  const int n0   = (wave >> 2) * 64;            // wave's 64-col patch
  const int lrow = lane & 15;
  const int koff = (lane >> 4) * 2;             // frag K sub-offset (lanes 16-31 hold K=2,3)

  v8f acc[2][4];
  {
    v8f vz = {};
    #pragma unroll
    for (int mi = 0; mi < 2; ++mi)
      #pragma unroll
      for (int ni = 0; ni < 4; ++ni) acc[mi][ni] = vz;
  }

  // async global->LDS mapping: 8 threads per 32-float row; 4 row-groups each
  const int grow = t >> 3;                      // 0..31
  const int gcol = (t & 7) * 4;                 // float4 column within K chunk
  const float* gA = abase + (size_t)grow * DDIM + gcol;
  const float* gB = bbase + (size_t)grow * DDIM + gcol;
  const unsigned la0 = as_lds_off(&lA[0][grow * LDST + gcol]);
  const unsigned lb0 = as_lds_off(&lB[0][grow * LDST + gcol]);
  const unsigned BUFBYTES = BUFELTS * 4u;
  const unsigned ROWSTEP  = 32u * LDST * 4u;    // LDS bytes per 32-row group

  auto prefetch = [&](int c) {
    const unsigned bsel = (unsigned)(c & 1) * BUFBYTES;
    const float* pa = gA + c * KC;
    const float* pb = gB + c * KC;
    #pragma unroll
    for (int r = 0; r < 4; ++r) {
      asm volatile("global_load_async_to_lds_b128 %0, %1, off"
                   :: "v"(la0 + bsel + (unsigned)r * ROWSTEP),
                      "v"((const void*)(pa + (size_t)r * 32 * DDIM))
                   : "memory");
      asm volatile("global_load_async_to_lds_b128 %0, %1, off"
                   :: "v"(lb0 + bsel + (unsigned)r * ROWSTEP),
                      "v"((const void*)(pb + (size_t)r * 32 * DDIM))
                   : "memory");
    }
  };

  prefetch(0);
  for (int c = 0; c < NCHUNK; ++c) {
    asm volatile("s_wait_asynccnt 0x0" ::: "memory");  // own chunk-c loads landed
    __syncthreads();                                   // everyone's landed; other buffer free
    if (c + 1 < NCHUNK) prefetch(c + 1);               // overlap next K-slab with WMMA below
    const float* A = lA[c & 1];
    const float* B = lB[c & 1];
    #pragma unroll
    for (int k = 0; k < KC; k += 4) {
      v2f af[2], bf[4];
      #pragma unroll
      for (int mi = 0; mi < 2; ++mi)
        af[mi] = *(const v2f*)(&A[(m0 + mi * 16 + lrow) * LDST + k + koff]);
      #pragma unroll
      for (int ni = 0; ni < 4; ++ni)
        bf[ni] = *(const v2f*)(&B[(n0 + ni * 16 + lrow) * LDST + k + koff]);
      #pragma unroll
      for (int mi = 0; mi < 2; ++mi)
        #pragma unroll
        for (int ni = 0; ni < 4; ++ni)
          acc[mi][ni] = __builtin_amdgcn_wmma_f32_16x16x4_f32(
              false, af[mi], false, bf[ni], (short)0, acc[mi][ni], false, false);
    }
  }

  __syncthreads();
  if (t < BM) sqI[t] = sq[i0 + t];
  else        sqJ[t - BM] = sq[j0 + t - BM];
  __syncthreads();

  const float g = gptr[0];
  const int halfsel = (lane >> 4) * 8;          // C layout: lanes 16-31 hold M = 8 + vgpr
  float local = 0.f;
  #pragma unroll
  for (int mi = 0; mi < 2; ++mi) {
    #pragma unroll
    for (int ni = 0; ni < 4; ++ni) {
      const float sj = sqJ[n0 + ni * 16 + lrow];
      #pragma unroll
      for (int e = 0; e < 8; ++e) {
        const float si = sqI[m0 + mi * 16 + e + halfsel];
        const float l2 = si + sj - 2.f * acc[mi][ni][e];
        const float t1 = __expf(-l2 * g);       // gamma_min term
        const float t2 = t1 * t1;
        const float t4 = t2 * t2;
        const float t8 = t4 * t4;
        const float t16 = t8 * t8;
        local += t1 + t2 + t4 + t8 + t16;       // all 5 RBF kernels
      }
    }
  }
  #pragma unroll
  for (int off = 16; off; off >>= 1) local += __shfl_xor(local, off, 32);
  if (lane == 0) redw[wave] = local;
  __syncthreads();
  if (t == 0) {
    float s = 0.f;
    #pragma unroll
    for (int w = 0; w < 8; ++w) s += redw[w];
    partial[blockIdx.x] = s;                    // deterministic: no atomics
  }
}

// ---------------- pass 4: signed quadrant combine (upper triangle only) ------
__global__ __launch_bounds__(256)
void mmd_finish(const float* __restrict__ partial, float* __restrict__ out) {
  __shared__ float red[256];
  const int t = threadIdx.x;
  float s = 0.f;
  for (int b = t; b < 4096; b += 256) {
    const int by = b >> 6, bx = b & 63;
    if (by > bx) continue;                                      // never written
    const float sign = ((by < 32) == (bx < 32)) ? 1.f : -1.f;   // XX,YY: +  XY,YX: -
    const float w = (by == bx) ? 1.f : 2.f;                     // transpose tile folded in
    s += sign * w * partial[b];
  }
  red[t] = s; __syncthreads();
  for (int off = 128; off; off >>= 1) { if (t < off) red[t] += red[t + off]; __syncthreads(); }
  if (t == 0) out[0] = red[0] * (1.f / ((float)N_HALF * (float)N_HALF));
}

extern "C" void kernel_launch(void* const* d_in, const int* in_sizes, int n_in,
                              void* d_out, int out_size, void* d_ws, size_t ws_size,
                              hipStream_t stream) {
  const float* src = (const float*)d_in[0];
  const float* tgt = (const float*)d_in[1];
  float* out = (float*)d_out;

  float* ws      = (float*)d_ws;
  float* colsum  = ws;            // 512
  float* gmin    = ws + 512;      // 1
  float* sq      = ws + 1024;     // 8192
  float* partial = ws + 10240;    // 4096

  mmd_colsum   <<<2,    256, 0, stream>>>(src, tgt, colsum);
  mmd_rowsq    <<<1024, 256, 0, stream>>>(src, tgt, sq);
  mmd_bandwidth<<<1,    256, 0, stream>>>(sq, colsum, gmin);
  mmd_gram     <<<4096, 256, 0, stream>>>(src, tgt, sq, gmin, partial);
  mmd_finish   <<<1,    256, 0, stream>>>(partial, out);
}